// PersLay_111669150108
// MI455X (gfx1250) — compile-verified
//
#include <hip/hip_runtime.h>

// ---- problem constants (from reference) ----
#define T_TYPES 7
#define B_GRAPHS 512
#define P_PTS 200
#define P_PAD 208            // pad to 13 tiles of 16 rows
#define Q_PTS 64
#define WW 64
#define WF 256
#define C_OUT 10
#define TQ (T_TYPES * Q_PTS) // 448

typedef __attribute__((ext_vector_type(2))) float v2f;
typedef __attribute__((ext_vector_type(8))) float v8f;

// D = A(16x4 f32) * B(4x16 f32) + C(16x16 f32), full fp32 WMMA
__device__ __forceinline__ v8f wmma_f32_4(v2f a, v2f b, v8f c) {
  return __builtin_amdgcn_wmma_f32_16x16x4_f32(
      /*neg_a=*/false, a, /*neg_b=*/false, b,
      /*c_mod=*/(short)0, c, /*reuse_a=*/false, /*reuse_b=*/false);
}

// =====================================================================
// Kernel A: per-(t,b) segment. Weight MLP (2->64->64->1) with WMMA for
// the 64x64 layer, then gaussian landmark features + weighted mean pool.
// Writes topo[b][t*Q + q].
// =====================================================================
__global__ __launch_bounds__(256) void perslay_pool_kernel(
    const float* __restrict__ diagms, const float* __restrict__ theta,
    const float* __restrict__ w1, const float* __restrict__ b1,
    const float* __restrict__ w2, const float* __restrict__ b2,
    const float* __restrict__ w3, const float* __restrict__ b3,
    float* __restrict__ topo)
{
  __shared__ float sP[P_PAD][2];          // points (padded w/ zeros)
  __shared__ float sH1[P_PAD][WW];        // relu(x@w1+b1)
  __shared__ float sW1[2][WW];
  __shared__ float sB1[WW], sB2[WW], sW3[WW];
  __shared__ float sThX[Q_PTS], sThY[Q_PTS];
  __shared__ float sWr[P_PAD];            // per-point sigmoid weight
  __shared__ float sPool[4][Q_PTS];
  __shared__ float sB3;

  const int tid = threadIdx.x;
  const int b = blockIdx.x;
  const int t = blockIdx.y;

  if (tid < WW) {
    sW1[0][tid] = w1[tid];
    sW1[1][tid] = w1[WW + tid];
    sB1[tid] = b1[tid];
    sB2[tid] = b2[tid];
    sW3[tid] = w3[tid];
    sThX[tid] = theta[2 * tid];
    sThY[tid] = theta[2 * tid + 1];
  }
  if (tid == 0) sB3 = b3[0];

  const float* dptr =
      diagms + ((size_t)t * (size_t)(B_GRAPHS * P_PTS) + (size_t)b * P_PTS) * 2;
  for (int i = tid; i < P_PAD; i += 256) {
    float x = 0.f, y = 0.f;
    if (i < P_PTS) { x = dptr[2 * i]; y = dptr[2 * i + 1]; }
    sP[i][0] = x; sP[i][1] = y;
  }
  __syncthreads();

  // h1 = relu(pts @ w1 + b1), rows >= P_PTS zeroed
  for (int i = tid; i < P_PAD * WW; i += 256) {
    int r = i >> 6, c = i & 63;
    float v = 0.f;
    if (r < P_PTS)
      v = fmaxf(sP[r][0] * sW1[0][c] + sP[r][1] * sW1[1][c] + sB1[c], 0.f);
    sH1[r][c] = v;
  }
  __syncthreads();

  // h2 = relu(h1 @ w2 + b2); w = sigmoid(h2 @ w3 + b3)  -- WMMA f32
  const int wv = tid >> 5, ln = tid & 31;
  const int m = ln & 15, half = ln >> 4;   // C layout: N=m, M = v + 8*half
  for (int tb = wv; tb < P_PAD / 16; tb += 8) {
    const int row0 = tb * 16;
    v8f c0, c1, c2, c3;
    {
      float b0 = sB2[m], bB = sB2[16 + m], bC = sB2[32 + m], bD = sB2[48 + m];
#pragma unroll
      for (int v = 0; v < 8; ++v) { c0[v] = b0; c1[v] = bB; c2[v] = bC; c3[v] = bD; }
    }
#pragma unroll
    for (int k0 = 0; k0 < WW; k0 += 4) {
      const int kk = k0 + half * 2;        // A: lanes 16-31 hold K=+2,+3
      v2f a; a.x = sH1[row0 + m][kk]; a.y = sH1[row0 + m][kk + 1];
      v2f bb;
      bb.x = w2[kk * WW + m];           bb.y = w2[(kk + 1) * WW + m];
      c0 = wmma_f32_4(a, bb, c0);
      bb.x = w2[kk * WW + 16 + m];      bb.y = w2[(kk + 1) * WW + 16 + m];
      c1 = wmma_f32_4(a, bb, c1);
      bb.x = w2[kk * WW + 32 + m];      bb.y = w2[(kk + 1) * WW + 32 + m];
      c2 = wmma_f32_4(a, bb, c2);
      bb.x = w2[kk * WW + 48 + m];      bb.y = w2[(kk + 1) * WW + 48 + m];
      c3 = wmma_f32_4(a, bb, c3);
    }
    // per-lane partial of relu(h2) . w3 along this lane's N columns
    float p[8];
#pragma unroll
    for (int v = 0; v < 8; ++v)
      p[v] = fmaxf(c0[v], 0.f) * sW3[m]      + fmaxf(c1[v], 0.f) * sW3[16 + m]
           + fmaxf(c2[v], 0.f) * sW3[32 + m] + fmaxf(c3[v], 0.f) * sW3[48 + m];
    // reduce over the 16 lanes of this half (xor masks stay in-half)
#pragma unroll
    for (int v = 0; v < 8; ++v) {
      p[v] += __shfl_xor(p[v], 1);
      p[v] += __shfl_xor(p[v], 2);
      p[v] += __shfl_xor(p[v], 4);
      p[v] += __shfl_xor(p[v], 8);
    }
#pragma unroll
    for (int v = 0; v < 8; ++v)
      if (m == v)
        sWr[row0 + half * 8 + v] = 1.f / (1.f + __expf(-(p[v] + sB3)));
  }
  __syncthreads();

  // weighted gaussian pooling: pooled[q] = mean_r w[r]*exp(-50*d2)
  const int q = tid & 63, g = tid >> 6;
  const float tx = sThX[q], ty = sThY[q];
  float acc = 0.f;
  for (int r = g; r < P_PTS; r += 4) {
    float dx = sP[r][0] - tx, dy = sP[r][1] - ty;
    acc += sWr[r] * __expf(-50.f * (dx * dx + dy * dy));
  }
  sPool[g][q] = acc;
  __syncthreads();
  if (tid < Q_PTS) {
    float s = (sPool[0][tid] + sPool[1][tid] + sPool[2][tid] + sPool[3][tid]) *
              (1.f / (float)P_PTS);
    topo[(size_t)b * TQ + t * Q_PTS + tid] = s;
  }
}

// =====================================================================
// Kernel B: out = relu(relu(topo@r1+rb1)@r2+rb2)@r3+rb3, 16 rows / block
// =====================================================================
__global__ __launch_bounds__(128) void perslay_rho_kernel(
    const float* __restrict__ topo,
    const float* __restrict__ r1, const float* __restrict__ rb1,
    const float* __restrict__ r2, const float* __restrict__ rb2,
    const float* __restrict__ r3, const float* __restrict__ rb3,
    float* __restrict__ out)
{
  __shared__ float sT[16][TQ];    // 28.0 KB
  __shared__ float sZ1[16][WF];   // 16 KB
  __shared__ float sZ2[16][WF];   // 16 KB

  const int tid = threadIdx.x;
  const int row0 = blockIdx.x * 16;

  for (int i = tid; i < 16 * TQ; i += 128) {
    int r = i / TQ, c = i - r * TQ;
    sT[r][c] = topo[(size_t)(row0 + r) * TQ + c];
  }
  __syncthreads();

  const int wv = tid >> 5, ln = tid & 31;
  const int m = ln & 15, half = ln >> 4;
  const int n0 = wv * 64;           // each wave owns 64 N columns

  // ---- GEMM1: z1 = relu(topo @ r1 + rb1), K = 448 ----
  {
    v8f c0, c1, c2, c3;
    {
      float b0 = rb1[n0 + m], bB = rb1[n0 + 16 + m];
      float bC = rb1[n0 + 32 + m], bD = rb1[n0 + 48 + m];
#pragma unroll
      for (int v = 0; v < 8; ++v) { c0[v] = b0; c1[v] = bB; c2[v] = bC; c3[v] = bD; }
    }
    for (int k0 = 0; k0 < TQ; k0 += 4) {
      const int kk = k0 + half * 2;
      v2f a; a.x = sT[m][kk]; a.y = sT[m][kk + 1];
      v2f bb;
      bb.x = r1[(size_t)kk * WF + n0 + m];
      bb.y = r1[(size_t)(kk + 1) * WF + n0 + m];
      c0 = wmma_f32_4(a, bb, c0);
      bb.x = r1[(size_t)kk * WF + n0 + 16 + m];
      bb.y = r1[(size_t)(kk + 1) * WF + n0 + 16 + m];
      c1 = wmma_f32_4(a, bb, c1);
      bb.x = r1[(size_t)kk * WF + n0 + 32 + m];
      bb.y = r1[(size_t)(kk + 1) * WF + n0 + 32 + m];
      c2 = wmma_f32_4(a, bb, c2);
      bb.x = r1[(size_t)kk * WF + n0 + 48 + m];
      bb.y = r1[(size_t)(kk + 1) * WF + n0 + 48 + m];
      c3 = wmma_f32_4(a, bb, c3);
    }
#pragma unroll
    for (int v = 0; v < 8; ++v) {
      const int mr = v + 8 * half;
      sZ1[mr][n0 + m]      = fmaxf(c0[v], 0.f);
      sZ1[mr][n0 + 16 + m] = fmaxf(c1[v], 0.f);
      sZ1[mr][n0 + 32 + m] = fmaxf(c2[v], 0.f);
      sZ1[mr][n0 + 48 + m] = fmaxf(c3[v], 0.f);
    }
  }
  __syncthreads();

  // ---- GEMM2: z2 = relu(z1 @ r2 + rb2), K = 256 ----
  {
    v8f c0, c1, c2, c3;
    {
      float b0 = rb2[n0 + m], bB = rb2[n0 + 16 + m];
      float bC = rb2[n0 + 32 + m], bD = rb2[n0 + 48 + m];
#pragma unroll
      for (int v = 0; v < 8; ++v) { c0[v] = b0; c1[v] = bB; c2[v] = bC; c3[v] = bD; }
    }
    for (int k0 = 0; k0 < WF; k0 += 4) {
      const int kk = k0 + half * 2;
      v2f a; a.x = sZ1[m][kk]; a.y = sZ1[m][kk + 1];
      v2f bb;
      bb.x = r2[(size_t)kk * WF + n0 + m];
      bb.y = r2[(size_t)(kk + 1) * WF + n0 + m];
      c0 = wmma_f32_4(a, bb, c0);
      bb.x = r2[(size_t)kk * WF + n0 + 16 + m];
      bb.y = r2[(size_t)(kk + 1) * WF + n0 + 16 + m];
      c1 = wmma_f32_4(a, bb, c1);
      bb.x = r2[(size_t)kk * WF + n0 + 32 + m];
      bb.y = r2[(size_t)(kk + 1) * WF + n0 + 32 + m];
      c2 = wmma_f32_4(a, bb, c2);
      bb.x = r2[(size_t)kk * WF + n0 + 48 + m];
      bb.y = r2[(size_t)(kk + 1) * WF + n0 + 48 + m];
      c3 = wmma_f32_4(a, bb, c3);
    }
#pragma unroll
    for (int v = 0; v < 8; ++v) {
      const int mr = v + 8 * half;
      sZ2[mr][n0 + m]      = fmaxf(c0[v], 0.f);
      sZ2[mr][n0 + 16 + m] = fmaxf(c1[v], 0.f);
      sZ2[mr][n0 + 32 + m] = fmaxf(c2[v], 0.f);
      sZ2[mr][n0 + 48 + m] = fmaxf(c3[v], 0.f);
    }
  }
  __syncthreads();

  // ---- tail: out = z2 @ r3 + rb3 (256 -> 10), scalar ----
  for (int o = tid; o < 16 * C_OUT; o += 128) {
    int r = o / C_OUT, cc = o - r * C_OUT;
    float acc = rb3[cc];
    for (int k = 0; k < WF; ++k) acc += sZ2[r][k] * r3[k * C_OUT + cc];
    out[(size_t)(row0 + r) * C_OUT + cc] = acc;
  }
}

extern "C" void kernel_launch(void* const* d_in, const int* in_sizes, int n_in,
                              void* d_out, int out_size, void* d_ws, size_t ws_size,
                              hipStream_t stream) {
  const float* diagms = (const float*)d_in[0];
  // d_in[1] = seg_ids: structurally n/P (equal contiguous segments) -> unused
  const float* theta = (const float*)d_in[2];
  const float* w1  = (const float*)d_in[3];
  const float* b1  = (const float*)d_in[4];
  const float* w2  = (const float*)d_in[5];
  const float* b2  = (const float*)d_in[6];
  const float* w3  = (const float*)d_in[7];
  const float* b3  = (const float*)d_in[8];
  const float* r1  = (const float*)d_in[9];
  const float* rb1 = (const float*)d_in[10];
  const float* r2  = (const float*)d_in[11];
  const float* rb2 = (const float*)d_in[12];
  const float* r3  = (const float*)d_in[13];
  const float* rb3 = (const float*)d_in[14];

  float* topo = (float*)d_ws;   // [B, T*Q] = 512*448 floats (~0.9 MB)
  float* out  = (float*)d_out;  // [B, C] float32

  perslay_pool_kernel<<<dim3(B_GRAPHS, T_TYPES), 256, 0, stream>>>(
      diagms, theta, w1, b1, w2, b2, w3, b3, topo);
  perslay_rho_kernel<<<B_GRAPHS / 16, 128, 0, stream>>>(
      topo, r1, rb1, r2, rb2, r3, rb3, out);
}